// RegLoss_90434831385107
// MI455X (gfx1250) — compile-verified
//
#include <hip/hip_runtime.h>
#include <math.h>

#define N_CH     151
#define N_EDGES  47
#define WAVES_PER_BLOCK 8
#define BLOCK_THREADS   256

// Skeleton edges (parent, child)
__constant__ unsigned char c_par[N_EDGES] = {
    0,1,2,3,1,5,6,8,8,8,8,8,9,10,11,13,14,15,17,18,19,21,22,23,
    25,26,27,29,29,29,29,29,30,31,32,34,35,36,38,39,40,42,43,44,46,47,48};
__constant__ unsigned char c_chd[N_EDGES] = {
    1,2,3,29,5,6,8,9,13,17,21,25,10,11,12,14,15,16,18,19,20,22,23,24,
    26,27,28,30,34,38,42,46,31,32,33,35,36,37,39,40,41,43,44,45,47,48,49};

typedef float v2f __attribute__((ext_vector_type(2)));
typedef float v8f __attribute__((ext_vector_type(8)));

// Wave32 all-lane sum via V_WMMA_F32_16X16X4_F32 with A = ones:
// D[m,n] = sum_k B[k,n]; summing D vgpr0 across one 16-lane group yields the
// total of all 64 B elements (partial in B vgpr0, 0 in vgpr1).
// EXEC must be all ones at the call site.
__device__ __forceinline__ float wave_sum_wmma(float x) {
    v2f a = {1.0f, 1.0f};
    v2f b = {x, 0.0f};
    v8f c = {};
    v8f d = __builtin_amdgcn_wmma_f32_16x16x4_f32(
        /*neg_a=*/false, a, /*neg_b=*/false, b,
        /*c_mod=*/(short)0, c, /*reuse_a=*/false, /*reuse_b=*/false);
    float s = d[0];
#pragma unroll
    for (int m = 1; m < 16; m <<= 1) s += __shfl_xor(s, m, 32);
    return s; // uniform across the wave (rows of D are identical)
}

__global__ void zero_acc_kernel(double* acc) {
    acc[0] = 0.0;
    acc[1] = 0.0;
}

__global__ __launch_bounds__(BLOCK_THREADS)
void reg_loss_main(const float* __restrict__ preds,
                   const float* __restrict__ targets,
                   double* __restrict__ acc,
                   int n_rows) {
    __shared__ float  sh_p[WAVES_PER_BLOCK][N_CH + 1];
    __shared__ float  sh_t[WAVES_PER_BLOCK][N_CH + 1];
    __shared__ double red_l1[WAVES_PER_BLOCK];
    __shared__ double red_vel[WAVES_PER_BLOCK];

    const int warp  = threadIdx.x >> 5;
    const int lane  = threadIdx.x & 31;
    const int gwave = blockIdx.x * WAVES_PER_BLOCK + warp;
    const int nwave = gridDim.x * WAVES_PER_BLOCK;

    float* sp = sh_p[warp];
    float* st = sh_t[warp];

    const float tiny = 1.17549435e-38f; // np.finfo(float32).tiny
    float l1_acc  = 0.0f;
    float vel_acc = 0.0f;

    // Per-lane edge assignment (hoisted out of the row loop).
    const int e0  = lane;                 // always < 47
    const int e1  = lane + 32;            // valid if < 47
    const bool has_e1 = (e1 < N_EDGES);
    const int a0 = 3 * (int)c_par[e0];
    const int b0 = 3 * (int)c_chd[e0];
    const int a1 = has_e1 ? 3 * (int)c_par[e1] : 0;
    const int b1 = has_e1 ? 3 * (int)c_chd[e1] : 0;

    for (int row = gwave; row < n_rows; row += nwave) {
        const long long base = (long long)row * N_CH;

        // Stage both rows to LDS; accumulate L1 on the fly.
#pragma unroll
        for (int i = 0; i < 5; ++i) {
            int j = lane + 32 * i;
            if (j < N_CH) {
                float t = targets[base + j];
                float p = preds[base + j];
                st[j] = t;
                sp[j] = p;
                // pm - tm: 0 where t==0, else p - t (branchless)
                float m = (t != 0.0f) ? 1.0f : 0.0f;
                l1_acc += fabsf(p - t) * m;
            }
        }
        // Wave-local: make this wave's DS stores visible to its own DS loads.
        asm volatile("s_wait_dscnt 0" ::: "memory");

        // Edge phase: lane e (and e+32) owns edge e.
        // Pass-1 LDS reads run only on lanes with a second edge (has_e1).
#pragma unroll
        for (int pass = 0; pass < 2; ++pass) {
            const int e  = (pass == 0) ? e0 : e1;
            const int a3 = (pass == 0) ? a0 : a1;
            const int b3 = (pass == 0) ? b0 : b1;
            if (pass == 1 && !has_e1) break; // uniform-per-lane trip handled below

            float tp[3], tc[3], pp[3], pc[3], mm[3];
#pragma unroll
            for (int c = 0; c < 3; ++c) {
                tp[c] = st[a3 + c];
                tc[c] = st[b3 + c];
                pp[c] = sp[a3 + c];
                pc[c] = sp[b3 + c];
                mm[c] = st[c * N_EDGES + e];   // mask source channel c*47+e
            }

            float pd[3], td[3];
            float pl2 = 0.0f, tl2 = 0.0f;
#pragma unroll
            for (int c = 0; c < 3; ++c) {
                float mp = (tp[c] != 0.0f) ? 1.0f : 0.0f;  // mask(parent ch)
                float mc = (tc[c] != 0.0f) ? 1.0f : 0.0f;  // mask(child ch)
                pd[c] = pp[c] * mp - pc[c] * mc;           // masked preds diff
                td[c] = tp[c] - tc[c];                     // targets*mask == targets
                pl2 += pd[c] * pd[c];
                tl2 += td[c] * td[c];
            }
            float rp = 1.0f / (sqrtf(pl2) + tiny);
            float rt = 1.0f / (sqrtf(tl2) + tiny);

            float contrib = 0.0f;
#pragma unroll
            for (int c = 0; c < 3; ++c) {
                float m  = (mm[c] != 0.0f) ? 1.0f : 0.0f;
                float dd = (pd[c] * rp - td[c] * rt) * m;
                contrib += dd * dd;
            }
            // Lanes without a second edge contribute 0 (uniform mask multiply).
            vel_acc += (pass == 0) ? contrib : (has_e1 ? contrib : 0.0f);
        }
    }

    // Loop trip count is wave-uniform -> EXEC all ones here (WMMA requirement).
    float wl = wave_sum_wmma(l1_acc);
    float wv = wave_sum_wmma(vel_acc);

    if (lane == 0) {
        red_l1[warp]  = (double)wl;
        red_vel[warp] = (double)wv;
    }
    __syncthreads();
    if (threadIdx.x == 0) {
        double tl = 0.0, tv = 0.0;
#pragma unroll
        for (int w = 0; w < WAVES_PER_BLOCK; ++w) { tl += red_l1[w]; tv += red_vel[w]; }
        atomicAdd(&acc[0], tl);   // global_atomic_add_f64
        atomicAdd(&acc[1], tv);
    }
}

__global__ void finalize_kernel(const double* __restrict__ acc,
                                float* __restrict__ out,
                                double inv_l1, double inv_vel) {
    out[0] = (float)(acc[0] * inv_l1 + 0.1 * acc[1] * inv_vel);
}

extern "C" void kernel_launch(void* const* d_in, const int* in_sizes, int n_in,
                              void* d_out, int out_size, void* d_ws, size_t ws_size,
                              hipStream_t stream) {
    const float* preds   = (const float*)d_in[0];
    const float* targets = (const float*)d_in[1];

    const long long total = (long long)in_sizes[0];     // B*T*151
    const int n_rows      = (int)(total / N_CH);        // B*T
    const double inv_l1   = 1.0 / (double)total;
    const double inv_vel  = 1.0 / ((double)n_rows * (double)(3 * N_EDGES));

    double* acc = (double*)d_ws;

    zero_acc_kernel<<<1, 1, 0, stream>>>(acc);

    // 1024 blocks * 8 waves = 8192 waves; 16 rows per wave, coalesced loads.
    reg_loss_main<<<1024, BLOCK_THREADS, 0, stream>>>(preds, targets, acc, n_rows);

    finalize_kernel<<<1, 1, 0, stream>>>(acc, (float*)d_out, inv_l1, inv_vel);
}